// SO3onS2_12927851560961
// MI455X (gfx1250) — compile-verified
//
#include <hip/hip_runtime.h>
#include <stdint.h>

// ---------------- problem constants ----------------
#define LMAXV   8
#define NL      9            // LMAX+1
#define NPAIR   81           // 9*9 raw floats per array per batch elem; also 81 full-m complex pairs
#define NOUT    249          // 215 FULL_MAP + 34 POWER_MAP columns
#define TB      128          // threads per block (4 waves) -> ~87KB LDS -> 3 blocks per 320KB WGP
#define FSTRIDE 170          // floats per per-lane full-F row in LDS (8B aligned, bank-conflict free)

struct Term  { float c; unsigned short ia, ib; };     // 8 B
struct Group {                                        // 16 B
  unsigned int   start;
  unsigned short nt;
  unsigned short fidx;     // pair index of conj(F[lo])_k, 0xFFFF => power/zero mode
  unsigned short outcol;
  unsigned short flags;    // bit0: flush column
  float          w;        // symmetry weight: 0 (odd-parity zero), 1 (m3==0), 2 (m3>0 pair)
};

#define GROUP_CAP 8192
#define TERM_BYTE_OFF 262144   // terms start 256KB into d_ws (after 64B hdr + groups)

// ---------------- CDNA5 async global->LDS (probe-verified: (as1 int*, as3 int*, imm, imm)) ----
#if defined(__gfx1250__) && __has_builtin(__builtin_amdgcn_global_load_async_to_lds_b32)
#define HAVE_ASYNC_LDS 1
typedef __attribute__((address_space(1))) int as1_int;
typedef __attribute__((address_space(3))) int as3_int;
#endif

// ---------------- device CG (Racah) in f64 ----------------
__device__ __attribute__((noinline)) double cg_coef(int j1,int m1,int j2,int m2,int j3,int m3,const double* f){
  if (m1 + m2 != m3) return 0.0;
  if (j3 < abs(j1 - j2) || j3 > j1 + j2) return 0.0;
  double pre = sqrt((2.0*j3+1.0) * f[j1+j2-j3]*f[j1-j2+j3]*f[-j1+j2+j3] / f[j1+j2+j3+1]);
  pre *= sqrt(f[j1+m1]*f[j1-m1]*f[j2+m2]*f[j2-m2]*f[j3+m3]*f[j3-m3]);
  int kmin = 0;
  if (j2-j3-m1 > kmin) kmin = j2-j3-m1;
  if (j1-j3+m2 > kmin) kmin = j1-j3+m2;
  int kmax = j1+j2-j3;
  if (j1-m1 < kmax) kmax = j1-m1;
  if (j2+m2 < kmax) kmax = j2+m2;
  if (kmax < kmin) return 0.0;
  double s = 0.0;
#pragma clang loop unroll(disable)
  for (int k = kmin; k <= kmax; ++k){
    double d = f[k]*f[j1+j2-j3-k]*f[j1-m1-k]*f[j2+m2-k]*f[j3-j2+m1+k]*f[j3-j1-m2+k];
    s += ((k & 1) ? -1.0 : 1.0) / d;
  }
  return pre * s;
}

// Emit groups for one (l1,l2,lo) triple.
// Symmetries used (real input field => F_l^{-m} = (-1)^m conj(F_l^m)):
//  (1) G_{-m} = (-1)^{m + l1+l2-lo} conj(G_m):
//      FULL odd  l1+l2+lo : beta == 0 exactly -> single zero-group
//      FULL even l1+l2+lo : beta = Re(G_0 F3_0*) + 2*sum_{m>0} Re(G_m F3_m*)
//      POWER (any parity) : sum_k |G_k|^2 = |G_0|^2 + 2*sum_{m>0} |G_m|^2
//  (2) l1==l2 with even (2*l1-lo): CG symmetric under m1<->m2 -> fold to m1<=m2, x2 off-diagonal.
__device__ __attribute__((noinline)) int emit_triple(int l1,int l2,int lo,int pw,int outcol,
                           Group* G,int& ng, Term* T,int& nt,
                           const double* f,int gcap,int tcap){
  const int odd = (l1 + l2 + lo) & 1;
  if (!pw && odd){
    if (ng < gcap){
      G[ng].start = (unsigned)nt; G[ng].nt = 0;
      G[ng].fidx = 0xFFFFu; G[ng].outcol = (unsigned short)outcol;
      G[ng].flags = 1u; G[ng].w = 0.0f;
      ++ng;
    }
    return outcol + 1;
  }
  const int fold = (l1 == l2) && !((2*l1 - lo) & 1);
#pragma clang loop unroll(disable)
  for (int m3 = 0; m3 <= lo; ++m3){
    int start = nt;
#pragma clang loop unroll(disable)
    for (int m1 = -l1; m1 <= l1; ++m1){
      int m2 = m3 - m1;
      if (m2 < -l2 || m2 > l2) continue;
      double mult = 1.0;
      if (fold){
        if (m1 > m2) continue;          // keep m1 <= m2 only
        if (m1 < m2) mult = 2.0;        // off-diagonal pair counted twice
      }
      double c = cg_coef(l1,m1,l2,m2,lo,m3,f) * mult;
      if (c != 0.0 && nt < tcap){
        T[nt].c  = (float)c;
        T[nt].ia = (unsigned short)(l1*l1 + l1 + m1);   // full-m pair index
        T[nt].ib = (unsigned short)(l2*l2 + l2 + m2);
        ++nt;
      }
    }
    if (ng < gcap){
      G[ng].start  = (unsigned)start;
      G[ng].nt     = (unsigned short)(nt - start);
      G[ng].fidx   = pw ? (unsigned short)0xFFFFu : (unsigned short)(lo*lo + lo + m3);
      G[ng].outcol = (unsigned short)outcol;
      G[ng].flags  = (m3 == lo) ? 1u : 0u;
      G[ng].w      = (m3 == 0) ? 1.0f : 2.0f;
      ++ng;
    }
  }
  return outcol + 1;
}

// One-wave kernel: regenerates the constant tables deterministically each call.
__global__ void gen_tables_kernel(Group* G, Term* T, int* hdr, int gcap, int tcap){
  if (threadIdx.x != 0 || blockIdx.x != 0) return;
  double f[26];
  f[0] = 1.0;
#pragma clang loop unroll(disable)
  for (int i = 1; i < 26; ++i) f[i] = f[i-1] * (double)i;
  int ng = 0, nt = 0, outcol = 0;
  // FULL_MAP
#pragma clang loop unroll(disable)
  for (int l1 = 0; l1 <= LMAXV; ++l1)
#pragma clang loop unroll(disable)
    for (int l2 = l1; l2 <= LMAXV; ++l2){
      int lomax = l1 + l2; if (lomax > LMAXV) lomax = LMAXV;
#pragma clang loop unroll(disable)
      for (int lo = abs(l1 - l2); lo <= lomax; ++lo)
        outcol = emit_triple(l1,l2,lo,0,outcol,G,ng,T,nt,f,gcap,tcap);
    }
  // POWER_MAP (mirrors the python enumeration exactly)
#pragma clang loop unroll(disable)
  for (int l2 = 2; l2 <= LMAXV; ++l2){
    outcol = emit_triple(1,l2,l2-1,1,outcol,G,ng,T,nt,f,gcap,tcap);
    if (l2 >= 4) outcol = emit_triple(1,l2,l2,1,outcol,G,ng,T,nt,f,gcap,tcap);
  }
#pragma clang loop unroll(disable)
  for (int l2 = 3; l2 <= LMAXV; ++l2){
    outcol = emit_triple(2,l2,l2-2,1,outcol,G,ng,T,nt,f,gcap,tcap);
    outcol = emit_triple(2,l2,l2-1,1,outcol,G,ng,T,nt,f,gcap,tcap);
  }
  outcol = emit_triple(3,3,2,1,outcol,G,ng,T,nt,f,gcap,tcap);
#pragma clang loop unroll(disable)
  for (int l2 = 4; l2 <= LMAXV; ++l2){
    outcol = emit_triple(3,l2,l2-3,1,outcol,G,ng,T,nt,f,gcap,tcap);
    if ((l2 & 1) && l2 >= 5){
      outcol = emit_triple(3,l2,l2-2,1,outcol,G,ng,T,nt,f,gcap,tcap);
      outcol = emit_triple(3,l2,l2-1,1,outcol,G,ng,T,nt,f,gcap,tcap);
    }
  }
  hdr[0] = ng; hdr[1] = nt; hdr[2] = outcol;  // outcol should equal NOUT (249)
}

// ---------------- main compute kernel ----------------
__global__ __launch_bounds__(TB) void bispectrum_kernel(
    const float* __restrict__ cre, const float* __restrict__ cim,
    float* __restrict__ out,
    const Group* __restrict__ G, const Term* __restrict__ T,
    const int* __restrict__ hdr, int B)
{
  extern __shared__ __align__(16) float smem[];
  const int tid = threadIdx.x;
  const long long bbase = (long long)blockIdx.x * TB;
  const int total = TB * NPAIR;                 // floats per staged array per block
  const long long glim = (long long)B * NPAIR;

  // ---- Stage 1: coalesced linear copy of the block's tile into LDS (CDNA5 async DMA) ----
  {
    const long long gbase = bbase * NPAIR;
    for (int f = tid; f < total; f += TB){
      long long gi = gbase + f;
      if (gi < glim){
#if defined(HAVE_ASYNC_LDS)
        __builtin_amdgcn_global_load_async_to_lds_b32(
            (as1_int*)(unsigned long long)(const void*)(cre + gi),
            (as3_int*)(unsigned)(unsigned long long)(void*)(smem + f), 0, 0);
        __builtin_amdgcn_global_load_async_to_lds_b32(
            (as1_int*)(unsigned long long)(const void*)(cim + gi),
            (as3_int*)(unsigned)(unsigned long long)(void*)(smem + total + f), 0, 0);
#else
        smem[f]         = cre[gi];
        smem[total + f] = cim[gi];
#endif
      }
    }
#if defined(HAVE_ASYNC_LDS)
#if __has_builtin(__builtin_amdgcn_s_wait_asynccnt)
    __builtin_amdgcn_s_wait_asynccnt(0);
#else
    asm volatile("s_wait_asynccnt 0" ::: "memory");
#endif
#endif
    __syncthreads();
  }

  // ---- Stage 2: each lane pulls its own row into registers (stride 81 odd -> conflict-free) ----
  float re[45], im[45];
  {
    const float* myR = smem + tid * NPAIR;
    const float* myI = smem + total + tid * NPAIR;
#pragma unroll
    for (int l = 0; l <= LMAXV; ++l){
#pragma unroll
      for (int m = 0; m <= l; ++m){
        re[l*(l+1)/2 + m] = myR[l*NL + m];
        im[l*(l+1)/2 + m] = myI[l*NL + m];
      }
    }
  }
  __syncthreads();   // all raw reads done before overwriting LDS with full-F layout

  // ---- Stage 3: materialize full-m F (F_l^{-m} = (-1)^m conj(F_l^m)) in own LDS row ----
  float* Frow = smem + (size_t)tid * FSTRIDE;   // 680 B rows: 8B aligned, 64-bank conflict free
#pragma unroll
  for (int l = 0; l <= LMAXV; ++l){
#pragma unroll
    for (int m = 0; m <= l; ++m){
      float r = re[l*(l+1)/2 + m];
      float i2 = im[l*(l+1)/2 + m];
      int p = l*l + l + m;
      Frow[2*p]   = r;
      Frow[2*p+1] = i2;
      if (m > 0){
        float s = (m & 1) ? -1.0f : 1.0f;
        int pn = l*l + l - m;
        Frow[2*pn]   = s * r;
        Frow[2*pn+1] = -s * i2;
      }
    }
  }
  // no barrier needed: below, each lane only reads its own row (in-order DS per wave)

  // ---- Stage 4: table-driven sparse trilinear evaluation (m3>=0 half-range) ----
  const long long b = bbase + tid;
  if (b >= B) return;
  const float2* Fc = (const float2*)Frow;
  float* orow = out + b * (long long)NOUT;
  const int ngroups = hdr[0];
  float acc = 0.0f;
  for (int g = 0; g < ngroups; ++g){
    Group gr = G[g];                       // uniform across lanes -> 1 request/wave
    const Term* tp = T + gr.start;
    const int n = gr.nt;
    __builtin_prefetch(tp + n, 0, 0);      // stream next group's terms (global_prefetch_b8)
    // unroll x2 with independent accumulators: breaks FMA chain, halves loop overhead
    float gre0 = 0.0f, gim0 = 0.0f, gre1 = 0.0f, gim1 = 0.0f;
    int t = 0;
    for (; t + 1 < n; t += 2){
      Term t0 = tp[t];
      Term t1 = tp[t+1];
      float2 a0 = Fc[t0.ia];
      float2 b0 = Fc[t0.ib];
      float2 a1 = Fc[t1.ia];
      float2 b1 = Fc[t1.ib];
      gre0 = fmaf(t0.c, fmaf(a0.x, b0.x, -(a0.y * b0.y)), gre0);
      gim0 = fmaf(t0.c, fmaf(a0.x, b0.y,  (a0.y * b0.x)), gim0);
      gre1 = fmaf(t1.c, fmaf(a1.x, b1.x, -(a1.y * b1.y)), gre1);
      gim1 = fmaf(t1.c, fmaf(a1.x, b1.y,  (a1.y * b1.x)), gim1);
    }
    if (t < n){
      Term t0 = tp[t];
      float2 a0 = Fc[t0.ia];
      float2 b0 = Fc[t0.ib];
      gre0 = fmaf(t0.c, fmaf(a0.x, b0.x, -(a0.y * b0.y)), gre0);
      gim0 = fmaf(t0.c, fmaf(a0.x, b0.y,  (a0.y * b0.x)), gim0);
    }
    const float gre = gre0 + gre1;
    const float gim = gim0 + gim1;
    float v;
    if (gr.fidx != 0xFFFFu){               // beta piece: Re(G_m * conj(F[lo]_m))
      float2 c3 = Fc[gr.fidx];
      v = fmaf(gre, c3.x, gim * c3.y);
    } else {                               // power piece: |G_m|^2  (or zero-group, w=0)
      v = fmaf(gre, gre, gim * gim);
    }
    acc = fmaf(gr.w, v, acc);
    if (gr.flags & 1u){ orow[gr.outcol] = acc; acc = 0.0f; }
  }
}

// ---------------- launch ----------------
extern "C" void kernel_launch(void* const* d_in, const int* in_sizes, int n_in,
                              void* d_out, int out_size, void* d_ws, size_t ws_size,
                              hipStream_t stream){
  const float* cre = (const float*)d_in[0];
  const float* cim = (const float*)d_in[1];
  float* out = (float*)d_out;
  const int B = in_sizes[0] / (NL * NL);

  char* ws = (char*)d_ws;
  int*   hdr = (int*)ws;                                   // 64 B header
  Group* G   = (Group*)(ws + 64);                          // up to 8192 groups (128 KB)
  Term*  T   = (Term*)(ws + TERM_BYTE_OFF);                // terms after 256 KB
  int tcap = (ws_size > TERM_BYTE_OFF)
               ? (int)((ws_size - TERM_BYTE_OFF) / sizeof(Term)) : 0;

  gen_tables_kernel<<<1, 32, 0, stream>>>(G, T, hdr, GROUP_CAP, tcap);

  const int grid = (B + TB - 1) / TB;
  const size_t shmem = (size_t)TB * FSTRIDE * sizeof(float);   // 87,040 B -> 3 blocks/WGP
  bispectrum_kernel<<<grid, TB, shmem, stream>>>(cre, cim, out, G, T, hdr, B);
}